// NeighborhoodAttention2D_76647986364461
// MI455X (gfx1250) — compile-verified
//
#include <hip/hip_runtime.h>
#include <hip/hip_bf16.h>
#include <math.h>

typedef float v2f __attribute__((ext_vector_type(2)));
typedef float v8f __attribute__((ext_vector_type(8)));

#define BB 2
#define HH 56
#define WW 56
#define C_DIM 256
#define NHEADS 8
#define HD 32
#define KW 7
#define M_ROWS (BB * HH * WW)   // 6272
#define NQKV 768                // 512 (qk) + 256 (v)
#define LDS_STRIDE 260          // 256 + 4 pad -> conflict-free LDS reads
#define MT 64                   // M-rows per GEMM block (4 subtiles of 16)

#define TS 8                    // attention pixel tile 8x8
#define HALO 14                 // TS + KW - 1
#define KV_STRIDE 36            // HD + 4 pad (keeps 16B alignment, breaks bank repeat)

// ---------------------------------------------------------------------------
// Kernel 1: fused QKV projection  P = X @ [w_qk ; w_v]^T  via f32 WMMA.
// Block = 128 threads (4 waves). Block computes 64 rows x 64 cols.
// A (64 x 256) staged in LDS; each wave holds 4 accumulators so every
// B fragment feeds 4 WMMAs (4 independent accumulation chains).
// ---------------------------------------------------------------------------
__global__ __launch_bounds__(128) void qkv_gemm_kernel(
    const float* __restrict__ x, const float* __restrict__ w_qk,
    const float* __restrict__ w_v,
    float* __restrict__ q_ws, float* __restrict__ k_ws, float* __restrict__ v_ws)
{
    __shared__ float ldsA[MT * LDS_STRIDE];   // 66,560 B
    const int nGroups = NQKV / 64;            // 12
    const int mBlk = blockIdx.x / nGroups;    // 0..97
    const int nGroup = blockIdx.x % nGroups;
    const int t = threadIdx.x;

    const float* xrow = x + (size_t)mBlk * MT * C_DIM;
    #pragma unroll
    for (int it = 0; it < 32; ++it) {
        int chunk = t + it * 128;             // 0..4095 float4 chunks
        int row = chunk >> 6;
        int col4 = (chunk & 63) << 2;
        float4 vv = *(const float4*)(xrow + row * C_DIM + col4);
        *(float4*)(&ldsA[row * LDS_STRIDE + col4]) = vv;
    }
    __syncthreads();

    const int wave = t >> 5;
    const int lane = t & 31;
    const int half = lane >> 4;               // K-half select
    const int ml = lane & 15;                 // row (A) / col (B) within tile
    const int nTile = nGroup * 64 + wave * 16;
    const int n = nTile + ml;

    // 16-wide weight tile never straddles the 512 (qk|v) boundary
    const float* wrow = (nTile < 512) ? (w_qk + (size_t)n * C_DIM)
                                      : (w_v + (size_t)(n - 512) * C_DIM);
    __builtin_prefetch(wrow, 0, 1);

    v8f acc[4];
    #pragma unroll
    for (int s = 0; s < 4; ++s) acc[s] = (v8f){0.f,0.f,0.f,0.f,0.f,0.f,0.f,0.f};

    const float* aBase = &ldsA[ml * LDS_STRIDE + 2 * half];
    const float* bBase = wrow + 2 * half;
    #pragma unroll 4
    for (int kk = 0; kk < 64; ++kk) {
        v2f b;
        b.x = bBase[4 * kk];  b.y = bBase[4 * kk + 1];
        #pragma unroll
        for (int s = 0; s < 4; ++s) {
            v2f a;
            a.x = aBase[s * 16 * LDS_STRIDE + 4 * kk];
            a.y = aBase[s * 16 * LDS_STRIDE + 4 * kk + 1];
            acc[s] = __builtin_amdgcn_wmma_f32_16x16x4_f32(
                false, a, false, b, (short)0, acc[s], false, false);
        }
    }

    const float scale = 0.17677669529663687f; // 32^-0.5
    #pragma unroll
    for (int s = 0; s < 4; ++s) {
        #pragma unroll
        for (int r = 0; r < 8; ++r) {
            int row = s * 16 + r + 8 * half;
            size_t gr = (size_t)(mBlk * MT + row);
            float val = acc[s][r];
            if (n < 256)      q_ws[gr * C_DIM + n] = val * scale;
            else if (n < 512) k_ws[gr * C_DIM + (n - 256)] = val;
            else              v_ws[gr * C_DIM + (n - 512)] = val;
        }
    }
}

// ---------------------------------------------------------------------------
// Kernel 2: 7x7 neighborhood attention, LDS-tiled.
// Block = 64 threads = 8x8 pixel tile for one (batch, head).
// k/v halo (14x14 rows x 32 floats, one head slice) staged in LDS;
// 49x neighbor reuse is served from LDS instead of L2.
// Window is always fully in-bounds (clipped start) -> no masking.
// ---------------------------------------------------------------------------
__global__ __launch_bounds__(64) void natten_kernel(
    const float* __restrict__ q_ws, const float* __restrict__ k_ws,
    const float* __restrict__ v_ws, float* __restrict__ o_ws,
    float* __restrict__ attn_out)
{
    __shared__ float k_t[HALO * HALO * KV_STRIDE];   // 28,224 B
    __shared__ float v_t[HALO * HALO * KV_STRIDE];

    int blk = blockIdx.x;             // (b*NHEADS + head)*49 + tile
    int tile = blk % 49;
    int bh = blk / 49;
    int head = bh % NHEADS;
    int b = bh / NHEADS;
    int i0 = (tile / 7) * TS;
    int j0 = (tile % 7) * TS;
    int rbase = i0 - 3; rbase = rbase < 0 ? 0 : (rbase > HH - HALO ? HH - HALO : rbase);
    int cbase = j0 - 3; cbase = cbase < 0 ? 0 : (cbase > WW - HALO ? WW - HALO : cbase);

    const int tid = threadIdx.x;

    // Cooperative stage of k/v halo: 196 rows x 8 float4 chunks each.
    for (int idx = tid; idx < HALO * HALO * 8; idx += 64) {
        int p = idx >> 3;                       // 0..195 halo pixel
        int c4 = (idx & 7) << 2;
        int pr = rbase + p / HALO;
        int pc = cbase + p % HALO;
        size_t src = ((size_t)(b * (HH * WW) + pr * WW + pc)) * C_DIM + head * HD + c4;
        *(float4*)(&k_t[p * KV_STRIDE + c4]) = *(const float4*)(k_ws + src);
        *(float4*)(&v_t[p * KV_STRIDE + c4]) = *(const float4*)(v_ws + src);
    }
    __syncthreads();

    int li = tid >> 3;                          // 0..7
    int lj = tid & 7;
    int i = i0 + li, j = j0 + lj;
    int pix = b * (HH * WW) + i * WW + j;
    int sh = i - 3; sh = sh < 0 ? 0 : (sh > HH - KW ? HH - KW : sh);
    int sw = j - 3; sw = sw < 0 ? 0 : (sw > WW - KW ? WW - KW : sw);
    int lr = sh - rbase;                        // 0..7, lr+6 <= 13
    int lc = sw - cbase;

    const float* qp = q_ws + (size_t)pix * C_DIM + head * HD;
    float4 qreg[8];
    #pragma unroll
    for (int d4 = 0; d4 < 8; ++d4) qreg[d4] = *(const float4*)(qp + 4 * d4);

    float logits[KW * KW];
    float mmax = -1e30f;
    #pragma unroll
    for (int p = 0; p < KW; ++p) {
        #pragma unroll
        for (int qq = 0; qq < KW; ++qq) {
            const float* kp = &k_t[((lr + p) * HALO + (lc + qq)) * KV_STRIDE];
            float acc = 0.f;
            #pragma unroll
            for (int d4 = 0; d4 < 8; ++d4) {
                float4 kv = *(const float4*)(kp + 4 * d4);
                float4 qv = qreg[d4];
                acc += qv.x * kv.x + qv.y * kv.y + qv.z * kv.z + qv.w * kv.w;
            }
            logits[p * KW + qq] = acc;
            mmax = fmaxf(mmax, acc);
        }
    }

    float s = 0.f;
    float o[HD];
    #pragma unroll
    for (int d = 0; d < HD; ++d) o[d] = 0.f;

    #pragma unroll
    for (int p = 0; p < KW; ++p) {
        #pragma unroll
        for (int qq = 0; qq < KW; ++qq) {
            int nidx = p * KW + qq;
            float w = __expf(logits[nidx] - mmax);
            logits[nidx] = w;
            s += w;
            const float* vp = &v_t[((lr + p) * HALO + (lc + qq)) * KV_STRIDE];
            #pragma unroll
            for (int d4 = 0; d4 < 8; ++d4) {
                float4 vv = *(const float4*)(vp + 4 * d4);
                o[4 * d4 + 0] += w * vv.x;
                o[4 * d4 + 1] += w * vv.y;
                o[4 * d4 + 2] += w * vv.z;
                o[4 * d4 + 3] += w * vv.w;
            }
        }
    }

    float inv = 1.0f / s;
    float* op = o_ws + (size_t)pix * C_DIM + head * HD;
    #pragma unroll
    for (int d4 = 0; d4 < 8; ++d4) {
        float4 ov;
        ov.x = o[4 * d4 + 0] * inv; ov.y = o[4 * d4 + 1] * inv;
        ov.z = o[4 * d4 + 2] * inv; ov.w = o[4 * d4 + 3] * inv;
        *(float4*)(op + 4 * d4) = ov;
    }

    // attn output: (B, NH, H, W, 49)
    size_t abase = (((size_t)(b * NHEADS + head) * (HH * WW)) + (size_t)i * WW + j) * (KW * KW);
    #pragma unroll
    for (int nidx = 0; nidx < KW * KW; ++nidx)
        attn_out[abase + nidx] = logits[nidx] * inv;
}

// ---------------------------------------------------------------------------
// Kernel 3: output projection  out = O @ w_proj^T + b_proj  via f32 WMMA.
// Same 64-row / 4-accumulator structure as kernel 1.
// ---------------------------------------------------------------------------
__global__ __launch_bounds__(128) void proj_gemm_kernel(
    const float* __restrict__ o_ws, const float* __restrict__ w_proj,
    const float* __restrict__ b_proj, float* __restrict__ out)
{
    __shared__ float ldsA[MT * LDS_STRIDE];
    const int nGroups = C_DIM / 64;           // 4
    const int mBlk = blockIdx.x / nGroups;    // 0..97
    const int nGroup = blockIdx.x % nGroups;
    const int t = threadIdx.x;

    const float* orow = o_ws + (size_t)mBlk * MT * C_DIM;
    #pragma unroll
    for (int it = 0; it < 32; ++it) {
        int chunk = t + it * 128;
        int row = chunk >> 6;
        int col4 = (chunk & 63) << 2;
        float4 vv = *(const float4*)(orow + row * C_DIM + col4);
        *(float4*)(&ldsA[row * LDS_STRIDE + col4]) = vv;
    }
    __syncthreads();

    const int wave = t >> 5;
    const int lane = t & 31;
    const int half = lane >> 4;
    const int ml = lane & 15;
    const int n = nGroup * 64 + wave * 16 + ml;

    const float* wrow = w_proj + (size_t)n * C_DIM;
    __builtin_prefetch(wrow, 0, 1);

    v8f acc[4];
    #pragma unroll
    for (int s = 0; s < 4; ++s) acc[s] = (v8f){0.f,0.f,0.f,0.f,0.f,0.f,0.f,0.f};

    const float* aBase = &ldsA[ml * LDS_STRIDE + 2 * half];
    const float* bBase = wrow + 2 * half;
    #pragma unroll 4
    for (int kk = 0; kk < 64; ++kk) {
        v2f b;
        b.x = bBase[4 * kk];  b.y = bBase[4 * kk + 1];
        #pragma unroll
        for (int s = 0; s < 4; ++s) {
            v2f a;
            a.x = aBase[s * 16 * LDS_STRIDE + 4 * kk];
            a.y = aBase[s * 16 * LDS_STRIDE + 4 * kk + 1];
            acc[s] = __builtin_amdgcn_wmma_f32_16x16x4_f32(
                false, a, false, b, (short)0, acc[s], false, false);
        }
    }

    float bias = b_proj[n];
    #pragma unroll
    for (int s = 0; s < 4; ++s) {
        #pragma unroll
        for (int r = 0; r < 8; ++r) {
            int row = s * 16 + r + 8 * half;
            size_t gr = (size_t)(mBlk * MT + row);
            out[gr * C_DIM + n] = acc[s][r] + bias;
        }
    }
}

// ---------------------------------------------------------------------------
extern "C" void kernel_launch(void* const* d_in, const int* in_sizes, int n_in,
                              void* d_out, int out_size, void* d_ws, size_t ws_size,
                              hipStream_t stream)
{
    const float* x      = (const float*)d_in[0];
    const float* w_qk   = (const float*)d_in[1];
    const float* w_v    = (const float*)d_in[2];
    const float* w_proj = (const float*)d_in[3];
    const float* b_proj = (const float*)d_in[4];

    float* out  = (float*)d_out;                           // (B,H,W,C)
    float* attn = out + (size_t)M_ROWS * C_DIM;            // (B,NH,H,W,49)

    const size_t plane = (size_t)M_ROWS * C_DIM;
    float* q_ws = (float*)d_ws;
    float* k_ws = q_ws + plane;
    float* v_ws = k_ws + plane;
    float* o_ws = v_ws + plane;

    // QKV projection: 98 M-blocks x 12 N-groups
    qkv_gemm_kernel<<<98 * 12, 128, 0, stream>>>(x, w_qk, w_v, q_ws, k_ws, v_ws);
    // Neighborhood attention: (2 batches x 8 heads x 49 tiles) blocks of 64
    natten_kernel<<<BB * NHEADS * 49, 64, 0, stream>>>(q_ws, k_ws, v_ws, o_ws, attn);
    // Output projection: 98 M-blocks x 4 N-groups
    proj_gemm_kernel<<<98 * 4, 128, 0, stream>>>(o_ws, w_proj, b_proj, out);
}